// NovelDistanceLoss_2087354106592
// MI455X (gfx1250) — compile-verified
//
#include <hip/hip_runtime.h>
#include <math.h>

typedef __attribute__((ext_vector_type(2))) float v2f;
typedef __attribute__((ext_vector_type(8))) float v8f;

#define NR 64
#define DC 256
#define BZ 8192
#define EPSN 1e-12f
#define MARGIN 1.0f

// ---------------------------------------------------------------------------
// Kernel 1: rel_n[r, :] = rel_weight[r, :] / max(||rel_weight[r, :]||, eps)
// One block (256 threads) per relation row.
// ---------------------------------------------------------------------------
__global__ __launch_bounds__(DC) void rel_normalize_kernel(
    const float* __restrict__ rel, float* __restrict__ rel_n) {
  __shared__ float red[DC];
  const int row = blockIdx.x;
  const int t   = threadIdx.x;
  float x = rel[(size_t)row * DC + t];
  red[t] = x * x;
  __syncthreads();
  for (int s = DC / 2; s > 0; s >>= 1) {
    if (t < s) red[t] += red[t + s];
    __syncthreads();
  }
  float inv = 1.0f / fmaxf(sqrtf(red[0]), EPSN);
  rel_n[(size_t)row * DC + t] = x * inv;
}

// ---------------------------------------------------------------------------
// Kernel 2: per block of 128 wo-rows:
//   - row inverse norms of wo
//   - 8 waves, each computes a 16x64 tile of dot(wo, rel_n^T) with
//     V_WMMA_F32_16X16X4_F32 (K-loop of 64 steps, 4 N-tiles)
//   - distances -> LDS -> per-row masked min / pos pick -> block partial sum
// ---------------------------------------------------------------------------
__global__ __launch_bounds__(256) void hard_negative_loss_kernel(
    const float* __restrict__ wo, const float* __restrict__ rel_n,
    const int* __restrict__ in_y, float* __restrict__ partials) {
  __shared__ float s_part[256];
  __shared__ float s_inv[128];
  __shared__ float s_dist[128][NR + 4];  // +4 pad to dodge bank conflicts

  const int t        = threadIdx.x;
  const int blk      = blockIdx.x;
  const int row_base = blk * 128;

  // ---- 1) per-row sum of squares of wo (2 threads per row) ----
  {
    const int row  = t >> 1;
    const int half = t & 1;
    const float* p = wo + (size_t)(row_base + row) * DC + half * 128;
    float ss = 0.f;
#pragma unroll 8
    for (int j = 0; j < 128; ++j) {
      float x = p[j];
      ss += x * x;
    }
    s_part[t] = ss;
  }
  __syncthreads();
  if (t < 128) {
    float ss = s_part[2 * t] + s_part[2 * t + 1];
    s_inv[t] = 1.0f / fmaxf(sqrtf(ss), EPSN);
  }
  __syncthreads();

  // ---- 2) WMMA GEMM: wave w -> 16 rows x 64 relations ----
  const int wave      = t >> 5;       // 0..7
  const int lane      = t & 31;
  const int mrow_base = wave * 16;    // local M base within the 128-row tile
  const int lhalf     = lane >> 4;    // 0 -> K pair (0,1), 1 -> K pair (2,3)
  const int l15       = lane & 15;

  // ISA 32-bit A layout: VGPR0 = K0 (lanes 0-15) / K2 (lanes 16-31),
  //                      VGPR1 = K1 / K3.  B mirrors with N across lanes.
  const float* arow = wo    + (size_t)(row_base + mrow_base + l15) * DC + 2 * lhalf;
  const float* b0   = rel_n + (size_t)( 0 + l15) * DC + 2 * lhalf;
  const float* b1   = rel_n + (size_t)(16 + l15) * DC + 2 * lhalf;
  const float* b2   = rel_n + (size_t)(32 + l15) * DC + 2 * lhalf;
  const float* b3   = rel_n + (size_t)(48 + l15) * DC + 2 * lhalf;

  v8f acc0 = {}, acc1 = {}, acc2 = {}, acc3 = {};

#pragma unroll 4
  for (int kb = 0; kb < DC; kb += 4) {
    v2f a   = *(const v2f*)(arow + kb);
    v2f bb0 = *(const v2f*)(b0 + kb);
    v2f bb1 = *(const v2f*)(b1 + kb);
    v2f bb2 = *(const v2f*)(b2 + kb);
    v2f bb3 = *(const v2f*)(b3 + kb);
    acc0 = __builtin_amdgcn_wmma_f32_16x16x4_f32(false, a, false, bb0, (short)0, acc0, false, false);
    acc1 = __builtin_amdgcn_wmma_f32_16x16x4_f32(false, a, false, bb1, (short)0, acc1, false, false);
    acc2 = __builtin_amdgcn_wmma_f32_16x16x4_f32(false, a, false, bb2, (short)0, acc2, false, false);
    acc3 = __builtin_amdgcn_wmma_f32_16x16x4_f32(false, a, false, bb3, (short)0, acc3, false, false);
  }

  // ---- 3) dots -> distances, stash to LDS.
  // D layout: VGPR v, lanes 0-15: M=v, N=lane; lanes 16-31: M=v+8, N=lane-16.
#pragma unroll
  for (int v = 0; v < 8; ++v) {
    const int mloc  = mrow_base + v + 8 * lhalf;
    const float inv = s_inv[mloc];
    float d0 = acc0[v] * inv;
    float d1 = acc1[v] * inv;
    float d2 = acc2[v] * inv;
    float d3 = acc3[v] * inv;
    s_dist[mloc][ 0 + l15] = sqrtf(fmaxf(2.f - 2.f * d0, 0.f));
    s_dist[mloc][16 + l15] = sqrtf(fmaxf(2.f - 2.f * d1, 0.f));
    s_dist[mloc][32 + l15] = sqrtf(fmaxf(2.f - 2.f * d2, 0.f));
    s_dist[mloc][48 + l15] = sqrtf(fmaxf(2.f - 2.f * d3, 0.f));
  }
  __syncthreads();

  // ---- 4) per-row: pos distance + hardest negative (min over r != y) ----
  float val = 0.f;
  if (t < 128) {
    const int y  = in_y[row_base + t];
    float posd   = 0.f;
    float minneg = 1e30f;
#pragma unroll
    for (int r = 0; r < NR; ++r) {
      float d = s_dist[t][r];
      if (r == y) posd = d;
      else        minneg = fminf(minneg, d);
    }
    val = posd + MARGIN - minneg;
  }
  s_part[t] = val;
  __syncthreads();
  for (int s = 128; s > 0; s >>= 1) {
    if (t < s) s_part[t] += s_part[t + s];
    __syncthreads();
  }
  if (t == 0) partials[blk] = s_part[0];
}

// ---------------------------------------------------------------------------
// Kernel 3: deterministic final reduction of 64 block partials -> mean
// ---------------------------------------------------------------------------
__global__ __launch_bounds__(64) void finalize_kernel(
    const float* __restrict__ partials, float* __restrict__ out) {
  __shared__ float red[64];
  const int t = threadIdx.x;
  red[t] = partials[t];
  __syncthreads();
  for (int s = 32; s > 0; s >>= 1) {
    if (t < s) red[t] += red[t + s];
    __syncthreads();
  }
  if (t == 0) out[0] = red[0] * (1.0f / (float)BZ);
}

// ---------------------------------------------------------------------------
extern "C" void kernel_launch(void* const* d_in, const int* in_sizes, int n_in,
                              void* d_out, int out_size, void* d_ws, size_t ws_size,
                              hipStream_t stream) {
  const float* wo   = (const float*)d_in[0];   // [8192, 256] f32
  const float* rel  = (const float*)d_in[1];   // [64, 256]   f32
  const int*   in_y = (const int*)d_in[2];     // [8192]      int

  float* rel_n    = (float*)d_ws;              // 64*256 floats
  float* partials = rel_n + NR * DC;           // 64 floats

  rel_normalize_kernel<<<NR, DC, 0, stream>>>(rel, rel_n);
  hard_negative_loss_kernel<<<BZ / 128, 256, 0, stream>>>(wo, rel_n, in_y, partials);
  finalize_kernel<<<1, 64, 0, stream>>>(partials, (float*)d_out);
}